// MCALoss_20572893348480
// MI455X (gfx1250) — compile-verified
//
#include <hip/hip_runtime.h>
#include <math.h>
#include <stdint.h>

typedef __attribute__((ext_vector_type(2))) float v2f;
typedef __attribute__((ext_vector_type(8))) float v8f;
typedef unsigned int u32x4 __attribute__((ext_vector_type(4)));
typedef int          i32x4 __attribute__((ext_vector_type(4)));
typedef int          i32x8 __attribute__((ext_vector_type(8)));

#define B_TOTAL     4096
#define D_DIM       128
#define NUM_CLASSES 1000
#define K_CENTERS   8
#define C_DIM       8000          // NUM_CLASSES * K_CENTERS
#define ALPHA       16.0f
#define TOPK        32
#define BAND_ROWS   256           // rows per band (dist slab = 256*8000*4 = 8 MB)
#define N_TILE      64            // N columns per workgroup (4 waves x 16)
#define INF_F       3.0e38f

#define HAVE_TDM (__has_builtin(__builtin_amdgcn_tensor_load_to_lds) && \
                  __has_builtin(__builtin_amdgcn_s_wait_tensorcnt))

// ---------------------------------------------------------------------------
// TDM: DMA a 2D tile (rows x row_dwords f32, contiguous rows) from global
// memory into LDS. If pad128 is set, the TDM inserts 1 DWORD of padding after
// every 128 DWORDs stored -> LDS row stride of 129 DWORDs (bank-conflict pad).
// Descriptor layout per CDNA5 ISA ch.8 (D# groups 0/1; remaining groups zero
// for a 2D tensor). This toolchain exposes the 6-arg builtin:
//   (u32x4 g0, i32x8 g1, i32x4, i32x4, i32x8, i32 cpol)
// ---------------------------------------------------------------------------
#if HAVE_TDM
__device__ __forceinline__ void tdm_load_2d_f32(unsigned lds_byte_off,
                                                const void* gptr,
                                                unsigned rows,
                                                unsigned row_dwords,
                                                bool pad128)
{
    unsigned long long ga = (unsigned long long)(uintptr_t)gptr;

    u32x4 g0;
    g0[0] = 1u;                                        // count=1 (valid), user mode
    g0[1] = lds_byte_off;                              // lds_addr (bytes)
    g0[2] = (unsigned)(ga & 0xFFFFFFFFu);              // global_addr[31:0]
    g0[3] = (unsigned)((ga >> 32) & 0x01FFFFFFu)       // global_addr[56:32]
          | (2u << 30);                                // type = 2 ("image")

    i32x8 g1;
    g1[0] = (int)((2u << 16)                                      // data_size = 4B
          | (pad128 ? ((1u << 20) | (6u << 22)) : 0u));           // pad_enable, interval=128 DW, amount=1 DW
    g1[1] = (int)((row_dwords & 0xFFFFu) << 16);                  // tensor_dim0[15:0]
    g1[2] = (int)(((row_dwords >> 16) & 0xFFFFu)                  // tensor_dim0[31:16]
          | ((rows & 0xFFFFu) << 16));                            // tensor_dim1[15:0]
    g1[3] = (int)(((rows >> 16) & 0xFFFFu)                        // tensor_dim1[31:16]
          | ((row_dwords & 0xFFFFu) << 16));                      // tile_dim0
    g1[4] = (int)(rows & 0xFFFFu);                                // tile_dim1 (tile_dim2 = 0)
    g1[5] = (int)row_dwords;                                      // tensor_dim0_stride[31:0]
    g1[6] = 0;                                                    // stride hi / dim1_stride lo
    g1[7] = 0;

    i32x4 z4 = {0, 0, 0, 0};
    i32x8 z8 = {0, 0, 0, 0, 0, 0, 0, 0};
    __builtin_amdgcn_tensor_load_to_lds(g0, g1, z4, z4, z8, 0);
}
#endif

// ---------------------------------------------------------------------------
// Zero the scalar output (harness poisons it with 0xAA).
// ---------------------------------------------------------------------------
__global__ void init_out_kernel(float* out) { out[0] = 0.0f; }

// ---------------------------------------------------------------------------
// Squared row norms: one wave32 per row, lane loads one float4 (16B),
// wave-reduces with xor shuffles.
// ---------------------------------------------------------------------------
__global__ __launch_bounds__(128) void sqnorm_kernel(const float* __restrict__ X,
                                                     float* __restrict__ out,
                                                     int rows)
{
    int row  = blockIdx.x * 4 + (threadIdx.x >> 5);
    int lane = threadIdx.x & 31;
    if (row >= rows) return;
    const float4* p = (const float4*)(X + (size_t)row * D_DIM);
    float4 f = p[lane];                                  // 32 lanes * 4 = 128 elems
    float s = f.x * f.x + f.y * f.y + f.z * f.z + f.w * f.w;
#pragma unroll
    for (int o = 16; o > 0; o >>= 1) s += __shfl_xor(s, o, 32);
    if (lane == 0) out[row] = s;
}

// ---------------------------------------------------------------------------
// Band GEMM + distance epilogue.
// grid = (C/N_TILE, BAND_ROWS/16), block = 128 (4 waves).
// TDM stages A (16x128) and B (64x128) f32 tiles into padded LDS; each wave
// then computes a 16x16 tile of dist = xx + yy - 2 * X*Cen^T with 32 unrolled
// V_WMMA_F32_16X16X4_F32 issues over K=128.
// ---------------------------------------------------------------------------
__global__ __launch_bounds__(128) void gemm_dist_kernel(
    const float* __restrict__ Xb,     // [BAND_ROWS, D]  (band slice of inputs)
    const float* __restrict__ Cen,    // [C, D]
    const float* __restrict__ xxb,    // [BAND_ROWS]     (band slice of ||x||^2)
    const float* __restrict__ yy,     // [C]             (||center||^2)
    float* __restrict__ dist)         // [BAND_ROWS, C]
{
    __shared__ float As[16][D_DIM + 1];       // +1 pad == TDM pad(128 DW -> +1 DW)
    __shared__ float Bs[N_TILE][D_DIM + 1];

    const int tid    = threadIdx.x;
    const int m_base = blockIdx.y * 16;
    const int n_base = blockIdx.x * N_TILE;

#if HAVE_TDM
    if (tid < 32) {   // one wave issues both DMA descriptors (EXEC ignored by TDM)
        tdm_load_2d_f32((unsigned)(uintptr_t)&As[0][0],
                        Xb + (size_t)m_base * D_DIM, 16, D_DIM, true);
        tdm_load_2d_f32((unsigned)(uintptr_t)&Bs[0][0],
                        Cen + (size_t)n_base * D_DIM, N_TILE, D_DIM, true);
        __builtin_amdgcn_s_wait_tensorcnt(0);
    }
    __syncthreads();
#else
    for (int i = tid; i < 16 * D_DIM; i += 128) {
        int r = i >> 7, c = i & (D_DIM - 1);
        As[r][c] = Xb[(m_base + r) * D_DIM + c];
    }
    for (int i = tid; i < N_TILE * D_DIM; i += 128) {
        int r = i >> 7, c = i & (D_DIM - 1);
        Bs[r][c] = Cen[(size_t)(n_base + r) * D_DIM + c];
    }
    __syncthreads();
#endif

    const int lane  = tid & 31;
    const int wave  = tid >> 5;            // 0..3: which 16-col sub-tile
    const int mn    = lane & 15;           // M for A-frag, N for B-frag
    const int khalf = (lane >> 4) << 1;    // lanes 0-15 -> K=k0,k0+1 ; lanes 16-31 -> k0+2,k0+3

    v8f acc = {};
#pragma unroll
    for (int k0 = 0; k0 < D_DIM; k0 += 4) {
        v2f a, b;
        a.x = As[mn][k0 + khalf];
        a.y = As[mn][k0 + khalf + 1];
        b.x = Bs[wave * 16 + mn][k0 + khalf];
        b.y = Bs[wave * 16 + mn][k0 + khalf + 1];
        // (neg_a, A, neg_b, B, c_mod, C, reuse_a, reuse_b)
        acc = __builtin_amdgcn_wmma_f32_16x16x4_f32(false, a, false, b,
                                                    (short)0, acc, false, false);
    }

    // C/D layout: VGPR r, lanes 0-15 -> M=r ; lanes 16-31 -> M=r+8 ; N = lane&15.
    const int   gn = n_base + wave * 16 + mn;
    const float yv = yy[gn];
#pragma unroll
    for (int r = 0; r < 8; ++r) {
        int   M    = r + ((lane >> 4) << 3);
        float dval = xxb[m_base + M] + yv - 2.0f * acc[r];
        dist[(size_t)(m_base + M) * C_DIM + gn] = dval;
    }
}

// ---------------------------------------------------------------------------
// Per-row selection + loss. One 256-thread workgroup per row.
// TDM stages the row's 8000 distances into LDS (1x8000 tile, no pad), then:
// unmasked min -> positive gather/mask -> 32 rounds of block argmin ->
// loss term accumulated atomically.
// ---------------------------------------------------------------------------
__global__ __launch_bounds__(256) void select_loss_kernel(
    const float* __restrict__ dist,     // [BAND_ROWS, C]
    const int* __restrict__ targetsb,   // [BAND_ROWS]
    float* __restrict__ out)
{
    __shared__ float v[C_DIM];          // 32 KB
    __shared__ float redv[256];
    __shared__ int   redi[256];
    __shared__ float pos[K_CENTERS];
    __shared__ float negs[TOPK];
    __shared__ float s_min;

    const int tid = threadIdx.x;
    const int row = blockIdx.x;
    const float* drow = dist + (size_t)row * C_DIM;

#if HAVE_TDM
    if (tid < 32) {
        tdm_load_2d_f32((unsigned)(uintptr_t)&v[0], drow, 1, C_DIM, false);
        __builtin_amdgcn_s_wait_tensorcnt(0);
    }
    __syncthreads();
#else
    for (int i = tid; i < C_DIM; i += 256) v[i] = drow[i];
    __syncthreads();
#endif

    // min over the FULL distance row (reference uses unmasked min for base)
    float lm = INF_F;
    for (int i = tid; i < C_DIM; i += 256) lm = fminf(lm, v[i]);
    redv[tid] = lm;
    __syncthreads();
    for (int s = 128; s > 0; s >>= 1) {
        if (tid < s) redv[tid] = fminf(redv[tid], redv[tid + s]);
        __syncthreads();
    }
    if (tid == 0) s_min = redv[0];
    __syncthreads();

    // gather positives (contiguous block of 8 centers for the target class),
    // then mask them out of the negative pool
    const int t = targetsb[row];
    if (tid < K_CENTERS) pos[tid] = v[t * K_CENTERS + tid];
    __syncthreads();
    if (tid < K_CENTERS) v[t * K_CENTERS + tid] = INF_F;
    __syncthreads();

    // 32 rounds of block argmin -> 32 smallest negatives, ascending
    for (int r = 0; r < TOPK; ++r) {
        float bm = INF_F;
        int   bi = 0;
        for (int i = tid; i < C_DIM; i += 256) {
            float x = v[i];
            if (x < bm) { bm = x; bi = i; }
        }
        redv[tid] = bm;
        redi[tid] = bi;
        __syncthreads();
        for (int s = 128; s > 0; s >>= 1) {
            if (tid < s && redv[tid + s] < redv[tid]) {
                redv[tid] = redv[tid + s];
                redi[tid] = redi[tid + s];
            }
            __syncthreads();
        }
        if (tid == 0) {
            negs[r]    = redv[0];
            v[redi[0]] = INF_F;
        }
        __syncthreads();
    }

    if (tid == 0) {
        float base = 0.5f * (negs[TOPK - 1] + s_min);
        float pe = 0.0f, ne = 0.0f;
        for (int j = 0; j < K_CENTERS; ++j) pe += expf(-ALPHA * (pos[j] - base));
        for (int j = 0; j < TOPK; ++j)      ne += expf(-ALPHA * (negs[j] - base));
        float loss = -logf(pe / (pe + ne));
        atomicAdd(out, loss * (1.0f / (float)B_TOTAL));
    }
}

// ---------------------------------------------------------------------------
// Host launcher. Workspace layout (floats):
//   xx[4096] | yy[8000] | dist[BAND_ROWS * 8000]   (~8.25 MB total)
// 16 bands: GEMM(band) then SELECT(band), serialized on `stream` so the
// 8 MB dist slab is reused.
// ---------------------------------------------------------------------------
extern "C" void kernel_launch(void* const* d_in, const int* in_sizes, int n_in,
                              void* d_out, int out_size, void* d_ws, size_t ws_size,
                              hipStream_t stream)
{
    const float* inputs  = (const float*)d_in[0];   // [4096, 128]
    const float* centers = (const float*)d_in[1];   // [8000, 128]
    const int*   targets = (const int*)d_in[2];     // [4096]
    // d_in[3] = center_labels (implied by contiguous layout), d_in[4] = _mask (all ones)
    float* out = (float*)d_out;

    float* xx   = (float*)d_ws;          // 4096
    float* yy   = xx + B_TOTAL;          // 8000
    float* dist = yy + C_DIM;            // BAND_ROWS * C_DIM

    init_out_kernel<<<1, 1, 0, stream>>>(out);
    sqnorm_kernel<<<B_TOTAL / 4, 128, 0, stream>>>(inputs, xx, B_TOTAL);
    sqnorm_kernel<<<C_DIM / 4, 128, 0, stream>>>(centers, yy, C_DIM);

    const int nbands = B_TOTAL / BAND_ROWS;   // 16
    for (int band = 0; band < nbands; ++band) {
        gemm_dist_kernel<<<dim3(C_DIM / N_TILE, BAND_ROWS / 16), 128, 0, stream>>>(
            inputs + (size_t)band * BAND_ROWS * D_DIM, centers,
            xx + band * BAND_ROWS, yy, dist);
        select_loss_kernel<<<BAND_ROWS, 256, 0, stream>>>(
            dist, targets + band * BAND_ROWS, out);
    }
}